// DiffNet_3556232921468
// MI455X (gfx1250) — compile-verified
//
#include <hip/hip_runtime.h>
#include <hip/hip_bf16.h>

typedef __attribute__((ext_vector_type(2))) float v2f;
typedef __attribute__((ext_vector_type(8))) float v8f;

#define B_DIM  32
#define I_DIM  1024
#define H1_DIM 512
#define H2_DIM 512
#define O_DIM  256

// -------- per-row stats: t[b] = sum_i v[b,i], q[b] = sum_i v[b,i]^2 --------
__global__ __launch_bounds__(256)
void row_stats_kernel(const float* __restrict__ act, int ncols,
                      float* __restrict__ stats /* [2*B]: t[0..B), q[B..2B) */) {
  __shared__ float s_t[256];
  __shared__ float s_q[256];
  const int b = blockIdx.x;
  const float* row = act + (size_t)b * ncols;
  float t = 0.f, q = 0.f;
  for (int i = threadIdx.x; i < ncols; i += 256) {
    float v = row[i];
    t += v;
    q += v * v;
  }
  s_t[threadIdx.x] = t;
  s_q[threadIdx.x] = q;
  __syncthreads();
  for (int off = 128; off > 0; off >>= 1) {
    if ((int)threadIdx.x < off) {
      s_t[threadIdx.x] += s_t[threadIdx.x + off];
      s_q[threadIdx.x] += s_q[threadIdx.x + off];
    }
    __syncthreads();
  }
  if (threadIdx.x == 0) {
    stats[b]         = s_t[0];
    stats[B_DIM + b] = s_q[0];
  }
}

// -------- fused layer: z = vi@W^T, vj = relu(z+bias), out = vj + delta --------
// One wave (32 threads) per 16x16 output tile; fp32 WMMA, K-step 4.
// A-operand layout (32-bit 16x4):  lane m=lane&15 holds K = k0+2*half, k0+2*half+1
// B-operand layout (32-bit 4x16):  lane n=lane&15 holds K = k0+2*half, k0+2*half+1
// -> both operands are one contiguous b64 load per lane per step.
__global__ __launch_bounds__(32)
void layer_wmma_kernel(const float* __restrict__ act_in,   // [B, K]
                       const float* __restrict__ W,        // [N, K] row-major
                       const float* __restrict__ bias,     // [N]
                       const float* __restrict__ stats,    // [2*B]
                       const float* __restrict__ c1w,      // [8,3]
                       const float* __restrict__ c1b,      // [8]
                       const float* __restrict__ c2w,      // [8]  (conv2_w[0,:])
                       const float* __restrict__ c2b,      // [1]
                       const int*   __restrict__ batch_num,// [1]
                       float* __restrict__ act_out,        // [B, N]
                       int K, int N) {
  const int lane  = threadIdx.x;   // 0..31, EXEC all-1s (WMMA requirement)
  const int half  = lane >> 4;     // 0 | 1
  const int l15   = lane & 15;
  const int mtile = blockIdx.y;    // 0..B/16-1
  const int ntile = blockIdx.x;    // 0..N/16-1

  const int m = mtile * 16 + l15;  // A row held by this lane
  const int n = ntile * 16 + l15;  // B column held by this lane

  const float* arow = act_in + (size_t)m * K + 2 * half;
  const float* brow = W      + (size_t)n * K + 2 * half;

  v8f acc = {};
  for (int k0 = 0; k0 < K; k0 += 4) {
    v2f a = *(const v2f*)(arow + k0);   // global_load_b64
    v2f b = *(const v2f*)(brow + k0);   // global_load_b64
    // D = A(16x4,f32) x B(4x16,f32) + C ; 8-arg form:
    // (neg_a, A, neg_b, B, c_mod, C, reuse_a, reuse_b)
    acc = __builtin_amdgcn_wmma_f32_16x16x4_f32(
        false, a, false, b, (short)0, acc, false, false);
  }

  // Collapse the two 1x1 convs (3->8->1 channel MLP, affine) to 4 scalars.
  float a0 = 0.f, a1 = 0.f, a2 = 0.f, d = c2b[0];
  #pragma unroll
  for (int h = 0; h < 8; ++h) {
    const float w2 = c2w[h];
    a0 += w2 * c1w[h * 3 + 0];   // coeff of vi
    a1 += w2 * c1w[h * 3 + 1];   // coeff of W
    a2 += w2 * c1w[h * 3 + 2];   // coeff of vj
    d  += w2 * c1b[h];
  }
  const float scale = 0.1f / (float)batch_num[0];

  // C/D f32 layout: VGPR r, lane -> row (r + 8*half), col l15 of the tile.
  const int   nout = ntile * 16 + l15;
  const float bn   = bias[nout];
  #pragma unroll
  for (int r = 0; r < 8; ++r) {
    const int   mout  = mtile * 16 + r + 8 * half;
    const float wdotv = acc[r];                  // sum_k W[n,k]*vi[m,k] (= z - bias)
    const float z     = wdotv + bn;
    const float vj    = fmaxf(z, 0.f);
    const float t     = stats[mout];
    const float q     = stats[B_DIM + mout];
    const float delta = scale * (a0 * q + a1 * wdotv + (a2 * vj + d) * t);
    act_out[(size_t)mout * N + nout] = vj + delta;
  }
}

extern "C" void kernel_launch(void* const* d_in, const int* in_sizes, int n_in,
                              void* d_out, int out_size, void* d_ws, size_t ws_size,
                              hipStream_t stream) {
  (void)in_sizes; (void)n_in; (void)out_size; (void)ws_size;

  const float* x       = (const float*)d_in[0];   // [B, I]
  const float* fc1_w   = (const float*)d_in[1];   // [H1, I]
  const float* fc1_b   = (const float*)d_in[2];   // [H1]
  const float* fc2_w   = (const float*)d_in[3];   // [H2, H1]
  const float* fc2_b   = (const float*)d_in[4];   // [H2]
  const float* fc3_w   = (const float*)d_in[5];   // [O, H2]
  const float* fc3_b   = (const float*)d_in[6];   // [O]
  const float* conv1_w = (const float*)d_in[7];   // [8,3]
  const float* conv1_b = (const float*)d_in[8];   // [8]
  const float* conv2_w = (const float*)d_in[9];   // [1,8] -> flat [8]
  const float* conv2_b = (const float*)d_in[10];  // [1]
  const int*   bnum    = (const int*)d_in[11];    // [1]

  float* out = (float*)d_out;                     // [B, O]

  // workspace layout (floats)
  float* act1   = (float*)d_ws;                   // [B*H1]
  float* act2   = act1 + B_DIM * H1_DIM;          // [B*H2]
  float* stats0 = act2 + B_DIM * H2_DIM;          // [2*B]
  float* stats1 = stats0 + 2 * B_DIM;             // [2*B]
  float* stats2 = stats1 + 2 * B_DIM;             // [2*B]

  const dim3 wave(32);

  // layer 1: x [B,I] -> act1 [B,H1]
  row_stats_kernel<<<B_DIM, 256, 0, stream>>>(x, I_DIM, stats0);
  layer_wmma_kernel<<<dim3(H1_DIM / 16, B_DIM / 16), wave, 0, stream>>>(
      x, fc1_w, fc1_b, stats0, conv1_w, conv1_b, conv2_w, conv2_b, bnum,
      act1, I_DIM, H1_DIM);

  // layer 2: act1 [B,H1] -> act2 [B,H2]
  row_stats_kernel<<<B_DIM, 256, 0, stream>>>(act1, H1_DIM, stats1);
  layer_wmma_kernel<<<dim3(H2_DIM / 16, B_DIM / 16), wave, 0, stream>>>(
      act1, fc2_w, fc2_b, stats1, conv1_w, conv1_b, conv2_w, conv2_b, bnum,
      act2, H1_DIM, H2_DIM);

  // layer 3: act2 [B,H2] -> out [B,O]
  row_stats_kernel<<<B_DIM, 256, 0, stream>>>(act2, H2_DIM, stats2);
  layer_wmma_kernel<<<dim3(O_DIM / 16, B_DIM / 16), wave, 0, stream>>>(
      act2, fc3_w, fc3_b, stats2, conv1_w, conv1_b, conv2_w, conv2_b, bnum,
      out, H2_DIM, O_DIM);
}